// FaceCaps_89816356094571
// MI455X (gfx1250) — compile-verified
//
#include <hip/hip_runtime.h>
#include <hip/hip_bf16.h>
#include <math.h>

// Problem constants
#define R_TOT   12800
#define NCAPS   25
#define OCH     64
#define ICH     8
#define NB      8
#define BCO     (NB * NCAPS * OCH)   // 12800 output elements

// Kernel config: one wave per capsule, 32 routes per workgroup
#define R_BLK   32
#define NWAVES  NCAPS
#define TPB     (NWAVES * 32)        // 800 threads
#define NBLOCKS (R_TOT / R_BLK)      // 400 workgroups

typedef __attribute__((ext_vector_type(2))) float v2f;
typedef __attribute__((ext_vector_type(8))) float v8f;

// Fused routing pass.
// MODE 0: uniform weights (iteration 0 softmax of zeros): num[b,c,o] += sum_r u_hat
// MODE 1: given v_in and prior logits b_in (null => 0):
//         per route: a[b,c]   = sum_o u_hat[b,c,o] * v_in[b,c,o]
//                    bnew     = b_in[r,c] + mean_b a[b,c]   (stored to b_out)
//                    e        = exp(bnew)
//                    den[c]  += e ; num[b,c,o] += e * u_hat[b,c,o]
template <int MODE>
__global__ __launch_bounds__(TPB, 1)
void caps_pass(const float* __restrict__ W, const float* __restrict__ x,
               const float* __restrict__ v_in, const float* __restrict__ b_in,
               float* __restrict__ b_out, float* __restrict__ num,
               float* __restrict__ den)
{
    const int wave = (int)threadIdx.x >> 5;   // capsule index c
    const int lane = (int)threadIdx.x & 31;
    const int c    = wave;
    const int b    = lane & 15;               // batch column (valid if < NB)
    const int h    = lane >> 4;               // half: +8 on M, +2 on K
    const int r0   = (int)blockIdx.x * R_BLK;

    // Preload v values this lane needs: v[b, c, o = t*16 + 8*h + j]
    float vv[32];
    if (MODE == 1) {
#pragma unroll
        for (int t = 0; t < 4; ++t)
#pragma unroll
            for (int j = 0; j < 8; ++j) {
                int o = t * 16 + 8 * h + j;
                vv[t * 8 + j] = (b < NB) ? v_in[(b * NCAPS + c) * OCH + o] : 0.0f;
            }
    }

    v8f acc[4];
#pragma unroll
    for (int t = 0; t < 4; ++t) acc[t] = (v8f){};
    float den_acc = 0.0f;

    for (int rr = 0; rr < R_BLK; ++rr) {
        const int r = r0 + rr;

        // B matrix (4x16 f32): B[k,n] = x[b=n, r, i=k(+4)]; lane: n=b, k = 2h+{0,1}
        v2f bm0 = (v2f){}, bm1 = (v2f){};
        if (b < NB) {
            const float* xp = x + ((size_t)b * R_TOT + r) * ICH + 2 * h;
            bm0.x = xp[0]; bm0.y = xp[1];       // i = 2h, 2h+1
            bm1.x = xp[4]; bm1.y = xp[5];       // i = 4+2h, 4+2h+1
        }

        const float* Wr = W + (((size_t)r * NCAPS + c) * OCH) * ICH;
        v8f d[4];
#pragma unroll
        for (int t = 0; t < 4; ++t) {
            // A matrix (16x4 f32): lane row m = lane&15, k = 2h+{0,1}
            const float* Ap = Wr + (size_t)(t * 16 + (lane & 15)) * ICH + 2 * h;
            v2f a0; a0.x = Ap[0]; a0.y = Ap[1];
            v2f a1; a1.x = Ap[4]; a1.y = Ap[5];
            v8f z = (MODE == 0) ? acc[t] : (v8f){};
            z = __builtin_amdgcn_wmma_f32_16x16x4_f32(false, a0, false, bm0,
                                                      (short)0, z, false, false);
            z = __builtin_amdgcn_wmma_f32_16x16x4_f32(false, a1, false, bm1,
                                                      (short)0, z, false, false);
            if (MODE == 0) acc[t] = z; else d[t] = z;
        }

        if (MODE == 1) {
            // Agreement: per-lane partial dot over this lane's 32 (o) slots
            float apart = 0.0f;
#pragma unroll
            for (int t = 0; t < 4; ++t)
#pragma unroll
                for (int j = 0; j < 8; ++j)
                    apart += d[t][j] * vv[t * 8 + j];
            // lane L and L+16 share batch b: combine halves
            apart += __shfl_xor(apart, 16, 32);
            // sum over batches b=0..7 (lanes 0-7 / 16-23 hold valid values)
            float asum = apart;
            asum += __shfl_xor(asum, 1, 32);
            asum += __shfl_xor(asum, 2, 32);
            asum += __shfl_xor(asum, 4, 32);
            float bprev = (b_in != nullptr) ? b_in[(size_t)r * NCAPS + c] : 0.0f;
            float bnew  = bprev + asum * 0.125f;       // mean over batch
            bnew = __shfl(bnew, 0, 32);                // broadcast lane 0's value
            if (lane == 0) b_out[(size_t)r * NCAPS + c] = bnew;
            float e = __expf(bnew);
            den_acc += e;
#pragma unroll
            for (int t = 0; t < 4; ++t)
#pragma unroll
                for (int j = 0; j < 8; ++j)
                    acc[t][j] += e * d[t][j];
        }
    }

    // Flush per-WG partials
    if (MODE == 1 && lane == 0) atomicAdd(&den[c], den_acc);
    if (b < NB) {
#pragma unroll
        for (int t = 0; t < 4; ++t)
#pragma unroll
            for (int j = 0; j < 8; ++j) {
                int o = t * 16 + 8 * h + j;
                atomicAdd(&num[(b * NCAPS + c) * OCH + o], acc[t][j]);
            }
    }
}

// squash(s) = s^2 * s / ((1+s^2) * sqrt(s^2))  (faithful elementwise form)
// den == nullptr => uniform softmax weight 1/R_TOT; else s = num/den[c]
__global__ void squash_kernel(const float* __restrict__ num,
                              const float* __restrict__ den,
                              float* __restrict__ out)
{
    int i = blockIdx.x * blockDim.x + threadIdx.x;
    if (i >= BCO) return;
    int cc = (i >> 6) % NCAPS;
    float w = (den != nullptr) ? (1.0f / den[cc]) : (1.0f / (float)R_TOT);
    float s  = num[i] * w;
    float sq = s * s;
    out[i] = sq * s / ((1.0f + sq) * sqrtf(sq));
}

extern "C" void kernel_launch(void* const* d_in, const int* in_sizes, int n_in,
                              void* d_out, int out_size, void* d_ws, size_t ws_size,
                              hipStream_t stream)
{
    const float* x = (const float*)d_in[0];   // [8, 12800, 8]
    const float* W = (const float*)d_in[1];   // [12800, 25, 64, 8]
    float* out = (float*)d_out;               // [8, 25, 64] (trailing dim 1)

    float* ws   = (float*)d_ws;
    float* num0 = ws;                  // 12800
    float* num1 = num0 + BCO;          // 12800
    float* num2 = num1 + BCO;          // 12800
    float* den1 = num2 + BCO;          // 32 (25 used)
    float* den2 = den1 + 32;           // 32
    float* v0   = den2 + 32;           // 12800
    float* v1   = v0 + BCO;            // 12800
    float* b1   = v1 + BCO;            // 320000
    float* b2   = b1 + (size_t)R_TOT * NCAPS;  // 320000

    // Zero the atomic accumulation buffers (num0..den2) each call
    hipMemsetAsync(ws, 0, (3 * BCO + 64) * sizeof(float), stream);

    dim3 grid(NBLOCKS), block(TPB);
    dim3 sgrid((BCO + 255) / 256), sblock(256);

    // Iteration 0: uniform weights -> num0 = sum_r u_hat ; v0 = squash(num0/R)
    caps_pass<0><<<grid, block, 0, stream>>>(W, x, nullptr, nullptr, nullptr, num0, nullptr);
    squash_kernel<<<sgrid, sblock, 0, stream>>>(num0, nullptr, v0);

    // Iteration 1: agreement with v0, logits b1 = mean a; weighted sums num1/den1
    caps_pass<1><<<grid, block, 0, stream>>>(W, x, v0, nullptr, b1, num1, den1);
    squash_kernel<<<sgrid, sblock, 0, stream>>>(num1, den1, v1);

    // Iteration 2: agreement with v1, logits b2 = b1 + mean a; weighted sums num2/den2
    caps_pass<1><<<grid, block, 0, stream>>>(W, x, v1, b1, b2, num2, den2);
    squash_kernel<<<sgrid, sblock, 0, stream>>>(num2, den2, out);
}